// GraphUpsampling_69904887710045
// MI455X (gfx1250) — compile-verified
//
#include <hip/hip_runtime.h>
#include <hip/hip_bf16.h>

typedef float v2f __attribute__((ext_vector_type(2)));
typedef float v8f __attribute__((ext_vector_type(8)));

#define NCHILD 8192
#define NCH    32
#define PARENT 4096

// ---------------- pass 0: zero the colsum accumulator ----------------
__global__ void zero_kernel(float* __restrict__ colsum) {
    int i = blockIdx.x * blockDim.x + threadIdx.x;
    if (i < NCHILD) colsum[i] = 0.0f;
}

// ---------------- pass 1: column sums of A (coalesced, atomic partials) ----
// grid (32, 64): blockIdx.x -> 256-column group, blockIdx.y -> 128-row chunk
__global__ __launch_bounds__(256) void colsum_kernel(const float* __restrict__ A,
                                                     float* __restrict__ colsum) {
    const int j  = blockIdx.x * 256 + threadIdx.x;   // column
    const int r0 = blockIdx.y * 128;                 // first row of chunk
    const float* p = A + (size_t)r0 * NCHILD + j;
    float s = 0.0f;
#pragma unroll 8
    for (int i = 0; i < 128; ++i)
        s += p[(size_t)i * NCHILD];
    atomicAdd(&colsum[j], s);
}

// ---------------- pass 2: build B' transposed: Bt[c][j] = input[desc[j]*32+c]/colsum[j]
// idx = c*8192 + j  (j fastest -> coalesced Bt writes & desc/colsum reads)
__global__ __launch_bounds__(256) void gather_kernel(const float* __restrict__ input,
                                                     const int*   __restrict__ desc,
                                                     const float* __restrict__ colsum,
                                                     float* __restrict__ Bt) {
    int idx = blockIdx.x * 256 + threadIdx.x;        // 0 .. 8192*32-1
    int j = idx & (NCHILD - 1);
    int c = idx >> 13;
    float inv = 1.0f / colsum[j];
    int p = desc[j];
    Bt[idx] = input[p * NCH + c] * inv;
}

// ---------------- pass 3: out = A @ B'  via V_WMMA_F32_16X16X4_F32 ----------
// grid: 512 blocks (one 16-row output tile each), 256 threads = 8 waves.
// Wave w covers K in [w*1024, (w+1)*1024); accumulates both 16-col N-tiles.
__global__ __launch_bounds__(256) void gemm_kernel(const float* __restrict__ A,
                                                   const float* __restrict__ Bt,
                                                   float* __restrict__ out) {
    __shared__ float red[2 * 8 * 32 * 8];            // [tile][wave][lane][i] = 16 KB

    const int lane = threadIdx.x & 31;
    const int w    = threadIdx.x >> 5;               // wave id 0..7
    const int hi   = lane >> 4;                      // lane half
    const int r    = lane & 15;
    const int rowBase = blockIdx.x * 16;

    // A 16x4 fragment: lane holds A[M=r][K = kbase + 2*hi + {0,1}]
    const float* ap  = A  + (size_t)(rowBase + r) * NCHILD + w * 1024 + 2 * hi;
    // B 4x16 fragment from Bt (32 x 8192): lane holds B[K=kbase+2*hi+{0,1}][N=r]
    const float* b0p = Bt + (size_t)r        * NCHILD + w * 1024 + 2 * hi;
    const float* b1p = Bt + (size_t)(r + 16) * NCHILD + w * 1024 + 2 * hi;

    v8f acc0 = {};
    v8f acc1 = {};

#pragma unroll 4
    for (int k = 0; k < 1024; k += 4) {
        v2f a  = *(const v2f*)(ap  + k);
        v2f b0 = *(const v2f*)(b0p + k);
        v2f b1 = *(const v2f*)(b1p + k);
        acc0 = __builtin_amdgcn_wmma_f32_16x16x4_f32(false, a, false, b0,
                                                     (short)0, acc0, false, false);
        acc1 = __builtin_amdgcn_wmma_f32_16x16x4_f32(false, a, false, b1,
                                                     (short)0, acc1, false, false);
    }

    // cross-wave reduction of the 8 K-partials through LDS
    float* s0 = &red[((0 * 8 + w) * 32 + lane) * 8];
    float* s1 = &red[((1 * 8 + w) * 32 + lane) * 8];
#pragma unroll
    for (int i = 0; i < 8; ++i) { s0[i] = acc0[i]; s1[i] = acc1[i]; }
    __syncthreads();

    if (w < 2) {                                     // wave w sums + stores tile w
        v8f sum = {};
#pragma unroll
        for (int ww = 0; ww < 8; ++ww) {
            const float* p = &red[((w * 8 + ww) * 32 + lane) * 8];
#pragma unroll
            for (int i = 0; i < 8; ++i) sum[i] += p[i];
        }
        // D layout: vgpr i, lane -> M = i + 8*hi, N = r ; out is (8192,32) row-major
        const int col = w * 16 + r;
#pragma unroll
        for (int i = 0; i < 8; ++i)
            out[(size_t)(rowBase + i + 8 * hi) * NCH + col] = sum[i];
    }
}

extern "C" void kernel_launch(void* const* d_in, const int* in_sizes, int n_in,
                              void* d_out, int out_size, void* d_ws, size_t ws_size,
                              hipStream_t stream) {
    const float* input = (const float*)d_in[0];   // (1,32,4096) f32, flat
    const float* A     = (const float*)d_in[1];   // (8192,8192) f32
    const int*   desc  = (const int*)  d_in[2];   // (8192,) int
    float*       out   = (float*)d_out;           // flat (8192,32)

    float* colsum = (float*)d_ws;                 // 8192 f32
    float* Bt     = colsum + NCHILD;              // 32*8192 f32 (B' transposed)

    zero_kernel  <<<NCHILD / 256, 256, 0, stream>>>(colsum);
    colsum_kernel<<<dim3(NCHILD / 256, NCHILD / 128), 256, 0, stream>>>(A, colsum);
    gather_kernel<<<(NCHILD * NCH) / 256, 256, 0, stream>>>(input, desc, colsum, Bt);
    gemm_kernel  <<<NCHILD / 16, 256, 0, stream>>>(A, Bt, out);
}